// OutliersQLinearWindowFix_39092792328874
// MI455X (gfx1250) — compile-verified
//
#include <hip/hip_runtime.h>

typedef float v2f __attribute__((ext_vector_type(2)));
typedef float v8f __attribute__((ext_vector_type(8)));

#define IC   4096   // K
#define OC   4096   // N
#define WIN  8

#define BM   64
#define BN   128
#define BK   32
#define LDA  36     // padded LDS stride (floats); 36 fl = 144 B, multiple of 16 B
#define LDB  36

// Async global -> LDS copy of 16 bytes (one b128 per lane), tracked by ASYNCcnt.
__device__ __forceinline__ void async_copy16(const float* gptr, const float* lptr) {
  const unsigned int       lds_off = (unsigned int)(size_t)lptr;   // wave-relative LDS offset
  const unsigned long long gaddr   = (unsigned long long)(size_t)gptr;
  asm volatile("global_load_async_to_lds_b128 %0, %1, off"
               :: "v"(lds_off), "v"(gaddr)
               : "memory");
}

__device__ __forceinline__ void wait_async_all() {
  asm volatile("s_wait_asynccnt 0x0" ::: "memory");
}

// ---------------------------------------------------------------------------
// Kernel 1: binarize weight with per-8-window outlier restore.
// One block per output-channel row; row staged once into LDS via async DMA.
// ---------------------------------------------------------------------------
__global__ __launch_bounds__(256) void binarize_kernel(
    const float* __restrict__ w, float* __restrict__ wb) {
  const int row = blockIdx.x;
  const int t   = threadIdx.x;
  const float* wr = w + (size_t)row * IC;

  __shared__ float rowbuf[IC];   // 16 KB
  __shared__ float red[256];

  // stage the whole row into LDS: 1024 x b128, 4 per thread
#pragma unroll
  for (int i = 0; i < 4; ++i) {
    const int idx = (t + i * 256) << 2;   // float index, multiple of 4
    async_copy16(wr + idx, &rowbuf[idx]);
  }
  wait_async_all();
  __syncthreads();

  // mean over the row
  float s = 0.f;
  for (int i = t; i < IC; i += 256) s += rowbuf[i];
  red[t] = s;
  __syncthreads();
  for (int off = 128; off > 0; off >>= 1) {
    if (t < off) red[t] += red[t + off];
    __syncthreads();
  }
  const float mu = red[0] * (1.0f / IC);
  __syncthreads();

  // mean |w - mu| over the row
  s = 0.f;
  for (int i = t; i < IC; i += 256) s += fabsf(rowbuf[i] - mu);
  red[t] = s;
  __syncthreads();
  for (int off = 128; off > 0; off >>= 1) {
    if (t < off) red[t] += red[t + off];
    __syncthreads();
  }
  const float scale = red[0] * (1.0f / IC);

  // per-window: keep argmax|w| element full precision, binarize the rest
  float* wbr = wb + (size_t)row * IC;
  for (int wi = t; wi < IC / WIN; wi += 256) {
    const int base = wi * WIN;
    float v[WIN];
#pragma unroll
    for (int j = 0; j < WIN; ++j) v[j] = rowbuf[base + j];
    int amax = 0;
    float best = fabsf(v[0]);
#pragma unroll
    for (int j = 1; j < WIN; ++j) {
      float a = fabsf(v[j]);
      if (a > best) { best = a; amax = j; }  // strict '>' keeps first max
    }
#pragma unroll
    for (int j = 0; j < WIN; ++j) {
      const float wc = v[j] - mu;
      const float q  = (wc > 0.f) ? scale : ((wc < 0.f) ? -scale : 0.f);
      wbr[base + j] = (j == amax) ? v[j] : q;
    }
  }
}

// ---------------------------------------------------------------------------
// Kernel 2: out[M,N] = X[M,K] * Wb[N,K]^T + bias, fp32 via V_WMMA_F32_16X16X4_F32.
// 256 threads = 8 waves (2 M x 4 N); block tile 64x128, K-tile 32.
// Double-buffered LDS filled by GLOBAL_LOAD_ASYNC_TO_LDS_B128 (ASYNCcnt),
// overlapped with the WMMA stream.
// ---------------------------------------------------------------------------
__global__ __launch_bounds__(256) void wmma_gemm_kernel(
    const float* __restrict__ x, const float* __restrict__ wb,
    const float* __restrict__ bias, float* __restrict__ out) {
  __shared__ float As[2][BM * LDA];   // 2 x 9216 B
  __shared__ float Bs[2][BN * LDB];   // 2 x 18432 B

  const int t     = threadIdx.x;
  const int lane  = t & 31;
  const int wave  = t >> 5;
  const int wm    = wave & 1;   // 0..1 : M sub-tile of 32
  const int wn    = wave >> 1;  // 0..3 : N sub-tile of 32
  const int lidx  = lane & 15;  // row/col within 16
  const int lhalf = lane >> 4;  // 0/1 -> K offset +2 (A/B frags), M offset +8 (C)

  const int m_blk = blockIdx.y * BM;
  const int n_blk = blockIdx.x * BN;

  // per-thread staging coordinates (row, float4-col within the K-tile)
  int aR[2], aC[2], bR[4], bC[4];
#pragma unroll
  for (int i = 0; i < 2; ++i) {
    const int idx = t + i * 256;        // 0..511
    aR[i] = idx >> 3;                   // 0..63
    aC[i] = (idx & 7) << 2;             // 0,4,...,28
  }
#pragma unroll
  for (int i = 0; i < 4; ++i) {
    const int idx = t + i * 256;        // 0..1023
    bR[i] = idx >> 3;                   // 0..127
    bC[i] = (idx & 7) << 2;
  }

  auto stage = [&](int buf, int k0) {
#pragma unroll
    for (int i = 0; i < 2; ++i)
      async_copy16(x + (size_t)(m_blk + aR[i]) * IC + (k0 + aC[i]),
                   &As[buf][aR[i] * LDA + aC[i]]);
#pragma unroll
    for (int i = 0; i < 4; ++i)
      async_copy16(wb + (size_t)(n_blk + bR[i]) * IC + (k0 + bC[i]),
                   &Bs[buf][bR[i] * LDB + bC[i]]);
  };

  v8f acc[2][2] = {};

  stage(0, 0);                      // prologue fill of buffer 0
  int p = 0;

  for (int k0 = 0; k0 < IC; k0 += BK) {
    wait_async_all();               // my DMA into buf p is done
    __syncthreads();                // everyone's DMA done; everyone done reading buf p^1

    if (k0 + BK < IC) stage(p ^ 1, k0 + BK);   // overlap next fill with compute

    const float* Arow[2] = {
      &As[p][(wm * 32 +      lidx) * LDA],
      &As[p][(wm * 32 + 16 + lidx) * LDA],
    };
    const float* Brow[2] = {
      &Bs[p][(wn * 32 +      lidx) * LDB],
      &Bs[p][(wn * 32 + 16 + lidx) * LDB],
    };

#pragma unroll
    for (int kk = 0; kk < BK; kk += 4) {
      const int ko = kk + (lhalf << 1);   // lanes 16-31 take K+2/K+3
      v2f a[2], b[2];
      a[0] = *(const v2f*)(Arow[0] + ko);
      a[1] = *(const v2f*)(Arow[1] + ko);
      b[0] = *(const v2f*)(Brow[0] + ko);
      b[1] = *(const v2f*)(Brow[1] + ko);
#pragma unroll
      for (int mt = 0; mt < 2; ++mt)
#pragma unroll
        for (int nt = 0; nt < 2; ++nt)
          acc[mt][nt] = __builtin_amdgcn_wmma_f32_16x16x4_f32(
              false, a[mt], false, b[nt], (short)0, acc[mt][nt],
              false, false);
    }
    p ^= 1;
  }

  // epilogue: C layout = VGPR j: lanes 0-15 -> M=j, lanes 16-31 -> M=j+8
  const int rbase = m_blk + wm * 32;
  const int cbase = n_blk + wn * 32;
#pragma unroll
  for (int mt = 0; mt < 2; ++mt) {
#pragma unroll
    for (int nt = 0; nt < 2; ++nt) {
      const int col = cbase + nt * 16 + lidx;
      const float bv = bias[col];
#pragma unroll
      for (int j = 0; j < 8; ++j) {
        const int row = rbase + mt * 16 + (lhalf << 3) + j;
        out[(size_t)row * OC + col] = acc[mt][nt][j] + bv;
      }
    }
  }
}

// ---------------------------------------------------------------------------
extern "C" void kernel_launch(void* const* d_in, const int* in_sizes, int n_in,
                              void* d_out, int out_size, void* d_ws, size_t ws_size,
                              hipStream_t stream) {
  (void)n_in; (void)out_size; (void)ws_size;
  const float* x    = (const float*)d_in[0];
  const float* w    = (const float*)d_in[1];
  const float* bias = (const float*)d_in[2];
  float* out = (float*)d_out;
  float* wb  = (float*)d_ws;               // 4096*4096 fp32 = 64 MB scratch

  const int M = in_sizes[0] / IC;          // 8192 rows (batch*seq)

  binarize_kernel<<<OC, 256, 0, stream>>>(w, wb);

  dim3 grid(OC / BN, M / BM);              // (32, 128)
  wmma_gemm_kernel<<<grid, 256, 0, stream>>>(x, wb, bias, out);
}